// GraphGeometricMorphism_35820027249065
// MI455X (gfx1250) — compile-verified
//
#include <hip/hip_runtime.h>

// CDNA5 / gfx1250 fused graph-attention:
//   QKV projection (WMMA f32 16x16x4), flash-style masked softmax-attention
//   (WMMA QK^T + exp + WMMA P@V, fused so the 256MB score matrix is never
//   materialized; adj (256MB) is the only large HBM stream -> ~11us floor
//   at 23.3 TB/s), deterministic two-stage mean reduction, tiny MLP head.
// Round 2: 2x-unrolled j-loop with ping-pong LDS P buffers (overlaps the
//   LDS round-trip of tile t with the score pipeline of tile t+1), adj loads
//   hoisted ahead of the score WMMAs, non-temporal adj stream (one-touch,
//   keeps K/V resident in the 192MB L2), and global_prefetch one iteration
//   ahead on the adj stream.

typedef float v2f __attribute__((ext_vector_type(2)));
typedef float v8f __attribute__((ext_vector_type(8)));

#define NV 8192   // num vertices
#define DD 16     // feature dim
#define HH 8      // hidden
#define JT 512    // number of 16-wide column tiles (NV/16)

static __device__ __forceinline__ v8f wmma4(v2f a, v2f b, v8f c) {
  // D = A(16x4) * B(4x16) + C, fp32
  return __builtin_amdgcn_wmma_f32_16x16x4_f32(
      /*neg_a=*/false, a, /*neg_b=*/false, b,
      /*c_mod=*/(short)0, c, /*reuse_a=*/false, /*reuse_b=*/false);
}

// ---------------------------------------------------------------------------
// Kernel 1: Q/K/V = sections @ W{q,k,v} + b{q,k,v}
// grid = 64 blocks x 256 threads; each wave handles one 16-row tile.
// ---------------------------------------------------------------------------
__global__ void qkv_kernel(const float* __restrict__ sections,
                           const float* __restrict__ Wq, const float* __restrict__ bq,
                           const float* __restrict__ Wk, const float* __restrict__ bk,
                           const float* __restrict__ Wv, const float* __restrict__ bv,
                           float* __restrict__ Q, float* __restrict__ K,
                           float* __restrict__ V) {
  const int lane = threadIdx.x & 31;
  const int wave = threadIdx.x >> 5;
  const int half = lane >> 4;   // 0: lanes 0-15, 1: lanes 16-31
  const int ln   = lane & 15;
  const int rowbase = (blockIdx.x * 8 + wave) * 16;

  // A slices (sections tile, 16x4 per slice): lane ln holds row M=ln,
  // K = 4s + 2*half + {0,1} -> single b64 load per slice.
  v2f a[4];
#pragma unroll
  for (int s = 0; s < 4; ++s)
    a[s] = *(const v2f*)(sections + (size_t)(rowbase + ln) * DD + 4 * s + 2 * half);

  const float* Wm[3] = {Wq, Wk, Wv};
  const float* bm[3] = {bq, bk, bv};
  float*       Om[3] = {Q, K, V};

#pragma unroll
  for (int m = 0; m < 3; ++m) {
    const float bias = bm[m][ln];          // C[m][n] starts at b[n]
    v8f c;
#pragma unroll
    for (int v = 0; v < 8; ++v) c[v] = bias;
#pragma unroll
    for (int s = 0; s < 4; ++s) {
      const int k0 = 4 * s + 2 * half;     // B slice: B[k][n] = W[k][n]
      v2f b;
      b.x = Wm[m][(size_t)k0 * DD + ln];
      b.y = Wm[m][(size_t)(k0 + 1) * DD + ln];
      c = wmma4(a[s], b, c);
    }
    // C layout: VGPR v, lane -> (M = v + 8*half, N = ln)
#pragma unroll
    for (int v = 0; v < 8; ++v)
      Om[m][(size_t)(rowbase + v + 8 * half) * DD + ln] = c[v];
  }
}

// ---------------------------------------------------------------------------
// One 16x16 column tile of the fused attention pipeline.
// ---------------------------------------------------------------------------
static __device__ __forceinline__ void attn_tile(
    int jb, float* __restrict__ Pbuf, const v2f qa[4],
    const float* __restrict__ K, const float* __restrict__ V,
    const int* __restrict__ adj, int rowbase, int half, int ln,
    v8f& num, float* __restrict__ den) {
  // adj loads first (independent of the WMMAs -> in flight early).
  // Non-temporal: 256MB one-touch stream, keep L2 for K/V.
  int am[8];
#pragma unroll
  for (int v = 0; v < 8; ++v)
    am[v] = __builtin_nontemporal_load(
        adj + (size_t)(rowbase + v + 8 * half) * NV + jb + ln);

  // scores tile: C = sum_s A_s(Q) * B_s(K^T).  B = K^T of row-major K
  // makes the B load identical in shape to an A load of K.
  v8f c = {};
#pragma unroll
  for (int s = 0; s < 4; ++s) {
    const v2f kb = *(const v2f*)(K + (size_t)(jb + ln) * DD + 4 * s + 2 * half);
    c = wmma4(qa[s], kb, c);
  }

  // mask + exp in C layout; accumulate denominator; stage P into LDS.
#pragma unroll
  for (int v = 0; v < 8; ++v) {
    const int m = v + 8 * half;
    const float e = am[v] ? __expf(c[v] * 0.25f) : 0.0f;  // scale = 1/sqrt(16)
    den[v] += e;
    Pbuf[m * DD + ln] = e;
  }

  // num += P(16x16) @ V_tile(16x16): re-read P from LDS in A layout.
#pragma unroll
  for (int s = 0; s < 4; ++s) {
    const v2f pa = *(const v2f*)(Pbuf + ln * DD + 4 * s + 2 * half);
    const int k0 = jb + 4 * s + 2 * half;                 // B[k][n] = V[jb+k][n]
    v2f vb;
    vb.x = V[(size_t)k0 * DD + ln];
    vb.y = V[(size_t)(k0 + 1) * DD + ln];
    num = wmma4(pa, vb, num);
  }
}

// ---------------------------------------------------------------------------
// Kernel 2: fused masked attention for one 16-row tile per block.
// grid = 512 blocks x 256 threads (8 waves); waves stride over column tiles
// two at a time with ping-pong P buffers. Emits per-block column sums of
// attended rows into part[] (deterministic, no float atomics).
// ---------------------------------------------------------------------------
__global__ void __launch_bounds__(256)
attn_kernel(const float* __restrict__ Q,
            const float* __restrict__ K,
            const float* __restrict__ V,
            const int* __restrict__ adj,
            float* __restrict__ part) {
  __shared__ float lds[8 * 512 + 8 * 16];   // 2 P buffers per wave + den

  const int lane = threadIdx.x & 31;
  const int wave = threadIdx.x >> 5;
  const int half = lane >> 4;
  const int ln   = lane & 15;
  const int rowbase = blockIdx.x * 16;
  float* buf0 = lds + wave * 512;
  float* buf1 = buf0 + 256;

  // Q tile as A slices, resident for the whole column sweep.
  v2f qa[4];
#pragma unroll
  for (int s = 0; s < 4; ++s)
    qa[s] = *(const v2f*)(Q + (size_t)(rowbase + ln) * DD + 4 * s + 2 * half);

  v8f num = {};                 // attended partial, C layout
  float den[8];                 // per-lane partial row sums of exp
#pragma unroll
  for (int v = 0; v < 8; ++v) den[v] = 0.0f;

  for (int jt = wave; jt < JT; jt += 16) {
    // Prefetch next iteration's adj rows (speculative, no counter cost).
    if (jt + 16 < JT) {
#pragma unroll
      for (int v = 0; v < 8; ++v)
        __builtin_prefetch(
            adj + (size_t)(rowbase + v + 8 * half) * NV + (jt + 16) * 16 + ln, 0, 0);
    }
    attn_tile(jt * 16, buf0, qa, K, V, adj, rowbase, half, ln, num, den);
    // jt+8 <= wave + 504 <= 511: always in range.
    attn_tile((jt + 8) * 16, buf1, qa, K, V, adj, rowbase, half, ln, num, den);
  }

  // Reduce den across the 16 lanes of each half (rows M = v + 8*half).
#pragma unroll
  for (int m = 1; m <= 8; m <<= 1) {
#pragma unroll
    for (int v = 0; v < 8; ++v) den[v] += __shfl_xor(den[v], m, 32);
  }

  // Cross-wave reduction through LDS (reuse the P regions).
#pragma unroll
  for (int v = 0; v < 8; ++v) lds[wave * 512 + v * 32 + lane] = num[v];
  if (ln == 0) {
#pragma unroll
    for (int v = 0; v < 8; ++v) lds[8 * 512 + wave * 16 + v + 8 * half] = den[v];
  }
  __syncthreads();

  if (wave == 0) {
    float numT[8], denT[8];
#pragma unroll
    for (int v = 0; v < 8; ++v) { numT[v] = 0.0f; denT[v] = 0.0f; }
    for (int w = 0; w < 8; ++w) {
#pragma unroll
      for (int v = 0; v < 8; ++v) {
        numT[v] += lds[w * 512 + v * 32 + lane];
        denT[v] += lds[8 * 512 + w * 16 + v + 8 * half];
      }
    }
    // attended[m][d] = num/den ; column-sum over this block's 16 rows.
    float colsum = 0.0f;
#pragma unroll
    for (int v = 0; v < 8; ++v) colsum += numT[v] / denT[v];
    colsum += __shfl_xor(colsum, 16, 32);                 // combine both halves
    if (half == 0) part[(size_t)blockIdx.x * DD + ln] = colsum;
  }
}

// ---------------------------------------------------------------------------
// Kernel 3: reduce block partials -> global mean feature, run the MLP head.
// ---------------------------------------------------------------------------
__global__ void head_kernel(const float* __restrict__ part,
                            const float* __restrict__ W1, const float* __restrict__ b1,
                            const float* __restrict__ W2, const float* __restrict__ b2,
                            float* __restrict__ out) {
  __shared__ float g[DD];
  const int t = threadIdx.x;
  if (t < DD) {
    float s = 0.0f;
    for (int b = 0; b < JT; ++b) s += part[(size_t)b * DD + t];
    g[t] = s * (1.0f / (float)NV);
  }
  __syncthreads();
  if (t == 0) {
    float logit = b2[0];
#pragma unroll
    for (int h = 0; h < HH; ++h) {
      float acc = b1[h];
#pragma unroll
      for (int d = 0; d < DD; ++d) acc += g[d] * W1[d * HH + h];
      logit += fmaxf(acc, 0.0f) * W2[h];
    }
    out[0] = logit;
  }
}

// ---------------------------------------------------------------------------
extern "C" void kernel_launch(void* const* d_in, const int* in_sizes, int n_in,
                              void* d_out, int out_size, void* d_ws, size_t ws_size,
                              hipStream_t stream) {
  const float* sections = (const float*)d_in[0];
  const int*   adj      = (const int*)d_in[1];
  const float* Wq = (const float*)d_in[2];
  const float* bq = (const float*)d_in[3];
  const float* Wk = (const float*)d_in[4];
  const float* bk = (const float*)d_in[5];
  const float* Wv = (const float*)d_in[6];
  const float* bv = (const float*)d_in[7];
  const float* W1 = (const float*)d_in[8];
  const float* b1 = (const float*)d_in[9];
  const float* W2 = (const float*)d_in[10];
  const float* b2 = (const float*)d_in[11];

  float* ws   = (float*)d_ws;
  float* part = ws;                       // 512*16 per-block partials
  float* Q    = part + JT * DD;           // 8192*16
  float* K    = Q + (size_t)NV * DD;
  float* V    = K + (size_t)NV * DD;

  qkv_kernel<<<64, 256, 0, stream>>>(sections, Wq, bq, Wk, bk, Wv, bv, Q, K, V);
  attn_kernel<<<JT, 256, 0, stream>>>(Q, K, V, adj, part);
  head_kernel<<<1, 32, 0, stream>>>(part, W1, b1, W2, b2, (float*)d_out);
}